// MyLSTM_72249939853984
// MI455X (gfx1250) — compile-verified
//
#include <hip/hip_runtime.h>

// ---------------------------------------------------------------------------
// LSTM on MI455X (gfx1250), fp32 WMMA (V_WMMA_F32_16X16X4_F32).
//
// Phase 1 (optional, if ws_size allows 512MB):  gx[t,b,g] = x_t @ W_ih^T + b
//   -> one massively parallel WMMA GEMM (131072 waves), off the critical path.
// Phase 2: persistent scan kernel, 32 blocks x 256 threads (co-resident),
//   software grid barrier between timesteps (atomic counter + agent fences),
//   so the 512-step recurrence runs inside ONE kernel launch.
//   Each wave owns a 16(batch) x 16(hidden col) tile for ALL FOUR gates, so
//   the gate nonlinearity + (h,c) update is fully lane-local.
//
// fp32 WMMA fragment layout (ISA 7.12.2):
//   A 16x4  : lane%16 = M row; K = 2*(lane/16) + {0,1} in the two VGPRs
//   B 4x16  : lane%16 = N col; K striped identically
//   C/D 16x16: VGPR r -> M = r + 8*(lane/16); N = lane%16
// ---------------------------------------------------------------------------

typedef __attribute__((ext_vector_type(2))) float v2f;
typedef __attribute__((ext_vector_type(8))) float v8f;

#define T_STEPS 512
#define BATCH   64
#define IN_DIM  1024
#define HID     1024
#define SCAN_BLOCKS 32   // 256 waves: 4 batch tiles x 64 hidden-col tiles

__device__ __forceinline__ float sigmoid_f(float x) {
    return 1.0f / (1.0f + __expf(-x));
}
__device__ __forceinline__ float tanh_f(float x) {
    return 2.0f / (1.0f + __expf(-2.0f * x)) - 1.0f;   // single v_exp_f32
}

// Software grid barrier: monotonically increasing counter, no reset needed
// within one call (init kernel zeroes it each call).  Release side waits the
// wave's stores (s_wait_storecnt) and publishes at agent scope; acquire side
// invalidates stale WGP$/L0 lines (global_inv) before the next step's reads.
__device__ __forceinline__ void grid_barrier(unsigned* bar, unsigned target) {
    __threadfence();                      // release: stores visible at L2
    __syncthreads();
    if (threadIdx.x == 0) {
        __hip_atomic_fetch_add(bar, 1u, __ATOMIC_RELAXED, __HIP_MEMORY_SCOPE_AGENT);
        unsigned v;
        do {
            __builtin_amdgcn_s_sleep(1);
            v = __hip_atomic_load(bar, __ATOMIC_RELAXED, __HIP_MEMORY_SCOPE_AGENT);
        } while (v < target);
    }
    __syncthreads();
    __threadfence();                      // acquire: invalidate stale lines
}

// ---------------------------------------------------------------------------
// Phase 1: gx[t,b,g] = sum_i x[t,b,i] * W_ih[g,i] + b_ih[g] + b_hh[g]
// Grid: 16384 blocks x 256 threads = 131072 waves; wave -> (t, mt, nt).
// Each wave: 16x16 tile for all 4 gate groups (A fragment shared 4x).
// ---------------------------------------------------------------------------
__global__ __launch_bounds__(256)
void input_gemm_kernel(const float* __restrict__ xs,
                       const float* __restrict__ W_ih,
                       const float* __restrict__ b_ih,
                       const float* __restrict__ b_hh,
                       float* __restrict__ gx)
{
    const int lane = threadIdx.x & 31;
    const int wave = (blockIdx.x << 3) + (threadIdx.x >> 5);
    const int t    = wave >> 8;
    const int rem  = wave & 255;
    const int mt   = rem & 3;
    const int nt   = rem >> 2;
    const int m_base = mt << 4;
    const int col    = (nt << 4) + (lane & 15);
    const int k_off  = (lane >> 4) << 1;

    const float* x_row = xs + ((size_t)t * BATCH + m_base + (lane & 15)) * IN_DIM;
    const float* wi_i = W_ih + (size_t)(0 * HID + col) * IN_DIM;
    const float* wi_f = W_ih + (size_t)(1 * HID + col) * IN_DIM;
    const float* wi_g = W_ih + (size_t)(2 * HID + col) * IN_DIM;
    const float* wi_o = W_ih + (size_t)(3 * HID + col) * IN_DIM;

    v8f acc_i = {}, acc_f = {}, acc_g = {}, acc_o = {};
#pragma unroll 4
    for (int k = 0; k < IN_DIM; k += 4) {
        const int kk = k + k_off;
        v2f a  = *(const v2f*)(x_row + kk);
        v2f bi = *(const v2f*)(wi_i + kk);
        v2f bf = *(const v2f*)(wi_f + kk);
        v2f bg = *(const v2f*)(wi_g + kk);
        v2f bo = *(const v2f*)(wi_o + kk);
        acc_i = __builtin_amdgcn_wmma_f32_16x16x4_f32(false, a, false, bi, (short)0, acc_i, false, false);
        acc_f = __builtin_amdgcn_wmma_f32_16x16x4_f32(false, a, false, bf, (short)0, acc_f, false, false);
        acc_g = __builtin_amdgcn_wmma_f32_16x16x4_f32(false, a, false, bg, (short)0, acc_g, false, false);
        acc_o = __builtin_amdgcn_wmma_f32_16x16x4_f32(false, a, false, bo, (short)0, acc_o, false, false);
    }

    const float bias_i = b_ih[0 * HID + col] + b_hh[0 * HID + col];
    const float bias_f = b_ih[1 * HID + col] + b_hh[1 * HID + col];
    const float bias_g = b_ih[2 * HID + col] + b_hh[2 * HID + col];
    const float bias_o = b_ih[3 * HID + col] + b_hh[3 * HID + col];

    const int m_half = (lane >> 4) << 3;
#pragma unroll
    for (int r = 0; r < 8; ++r) {
        const int m = m_base + r + m_half;
        float* grow = gx + ((size_t)t * BATCH + m) * (4 * HID);
        grow[0 * HID + col] = acc_i[r] + bias_i;
        grow[1 * HID + col] = acc_f[r] + bias_f;
        grow[2 * HID + col] = acc_g[r] + bias_g;
        grow[3 * HID + col] = acc_o[r] + bias_o;
    }
}

// ---------------------------------------------------------------------------
// Phase 2: persistent recurrent scan.  Grid: SCAN_BLOCKS x 256 (co-resident).
// If gx != nullptr the input projection is precomputed; otherwise it is fused
// into the per-step K loop (branch is grid-uniform -> EXEC stays all-1s).
// ---------------------------------------------------------------------------
__global__ __launch_bounds__(256)
void lstm_scan_kernel(const float* __restrict__ xs,
                      const float* __restrict__ W_ih,
                      const float* __restrict__ W_hh,
                      const float* __restrict__ b_ih,
                      const float* __restrict__ b_hh,
                      float* __restrict__ hbuf0,
                      float* __restrict__ hbuf1,
                      float* __restrict__ c_buf,
                      float* __restrict__ ys,      // d_out: ys | hT | cT
                      const float* __restrict__ gx,
                      unsigned* __restrict__ bar)
{
    const int lane  = threadIdx.x & 31;
    const int wave  = (blockIdx.x << 3) + (threadIdx.x >> 5);   // 0..255
    const int mt    = wave & 3;
    const int nt    = wave >> 2;
    const int m_base = mt << 4;
    const int col    = (nt << 4) + (lane & 15);
    const int k_off  = (lane >> 4) << 1;
    const int m_half = (lane >> 4) << 3;

    const float* wh_i = W_hh + (size_t)(0 * HID + col) * HID;
    const float* wh_f = W_hh + (size_t)(1 * HID + col) * HID;
    const float* wh_g = W_hh + (size_t)(2 * HID + col) * HID;
    const float* wh_o = W_hh + (size_t)(3 * HID + col) * HID;

    const float* wi_i = W_ih + (size_t)(0 * HID + col) * IN_DIM;
    const float* wi_f = W_ih + (size_t)(1 * HID + col) * IN_DIM;
    const float* wi_g = W_ih + (size_t)(2 * HID + col) * IN_DIM;
    const float* wi_o = W_ih + (size_t)(3 * HID + col) * IN_DIM;

    const float bias_i = b_ih[0 * HID + col] + b_hh[0 * HID + col];
    const float bias_f = b_ih[1 * HID + col] + b_hh[1 * HID + col];
    const float bias_g = b_ih[2 * HID + col] + b_hh[2 * HID + col];
    const float bias_o = b_ih[3 * HID + col] + b_hh[3 * HID + col];

    float* out_tail = ys + (size_t)T_STEPS * BATCH * HID;

    for (int t = 0; t < T_STEPS; ++t) {
        const float* h_in  = (t & 1) ? hbuf1 : hbuf0;
        float*       h_out = (t & 1) ? hbuf0 : hbuf1;
        const float* h_row = h_in + (size_t)(m_base + (lane & 15)) * HID;

        v8f acc_i = {}, acc_f = {}, acc_g = {}, acc_o = {};

        if (gx == nullptr) {
            // Fused input projection (fallback when workspace is small).
            const float* x_row = xs + ((size_t)t * BATCH + m_base + (lane & 15)) * IN_DIM;
#pragma unroll 4
            for (int k = 0; k < IN_DIM; k += 4) {
                const int kk = k + k_off;
                v2f a  = *(const v2f*)(x_row + kk);
                v2f bi = *(const v2f*)(wi_i + kk);
                v2f bf = *(const v2f*)(wi_f + kk);
                v2f bg = *(const v2f*)(wi_g + kk);
                v2f bo = *(const v2f*)(wi_o + kk);
                acc_i = __builtin_amdgcn_wmma_f32_16x16x4_f32(false, a, false, bi, (short)0, acc_i, false, false);
                acc_f = __builtin_amdgcn_wmma_f32_16x16x4_f32(false, a, false, bf, (short)0, acc_f, false, false);
                acc_g = __builtin_amdgcn_wmma_f32_16x16x4_f32(false, a, false, bg, (short)0, acc_g, false, false);
                acc_o = __builtin_amdgcn_wmma_f32_16x16x4_f32(false, a, false, bo, (short)0, acc_o, false, false);
            }
        }

        // Hidden-to-hidden contribution, K = HID over h_{t-1}.
#pragma unroll 4
        for (int k = 0; k < HID; k += 4) {
            const int kk = k + k_off;
            v2f a  = *(const v2f*)(h_row + kk);
            v2f bi = *(const v2f*)(wh_i + kk);
            v2f bf = *(const v2f*)(wh_f + kk);
            v2f bg = *(const v2f*)(wh_g + kk);
            v2f bo = *(const v2f*)(wh_o + kk);
            acc_i = __builtin_amdgcn_wmma_f32_16x16x4_f32(false, a, false, bi, (short)0, acc_i, false, false);
            acc_f = __builtin_amdgcn_wmma_f32_16x16x4_f32(false, a, false, bf, (short)0, acc_f, false, false);
            acc_g = __builtin_amdgcn_wmma_f32_16x16x4_f32(false, a, false, bg, (short)0, acc_g, false, false);
            acc_o = __builtin_amdgcn_wmma_f32_16x16x4_f32(false, a, false, bo, (short)0, acc_o, false, false);
        }

        // Gate nonlinearities + state update, fully lane-local.
        float* ys_t = ys + (size_t)t * BATCH * HID;
#pragma unroll
        for (int r = 0; r < 8; ++r) {
            const int m = m_base + r + m_half;
            const size_t idx = (size_t)m * HID + col;

            float pi, pf, pg, po;
            if (gx != nullptr) {
                const float* grow = gx + ((size_t)t * BATCH + m) * (4 * HID);
                pi = grow[0 * HID + col];
                pf = grow[1 * HID + col];
                pg = grow[2 * HID + col];
                po = grow[3 * HID + col];
            } else {
                pi = bias_i; pf = bias_f; pg = bias_g; po = bias_o;
            }

            const float gi = sigmoid_f(acc_i[r] + pi);
            const float gf = sigmoid_f(acc_f[r] + pf);
            const float gg = tanh_f   (acc_g[r] + pg);
            const float go = sigmoid_f(acc_o[r] + po);
            const float c_old = c_buf[idx];
            const float c_new = gf * c_old + gi * gg;
            const float h_new = go * tanh_f(c_new);
            c_buf[idx] = c_new;
            h_out[idx] = h_new;
            ys_t[idx]  = h_new;
            if (t == T_STEPS - 1) {                  // final (hT, cT)
                out_tail[idx]                = h_new;
                out_tail[BATCH * HID + idx]  = c_new;
            }
        }

        grid_barrier(bar, (unsigned)SCAN_BLOCKS * (unsigned)(t + 1));
    }
}

// ---------------------------------------------------------------------------
// Init: h0 -> ping buffer 0, c0 -> c buffer, zero the barrier counter.
// ---------------------------------------------------------------------------
__global__ void init_state_kernel(const float* __restrict__ h0,
                                  const float* __restrict__ c0,
                                  float* __restrict__ hbuf0,
                                  float* __restrict__ cbuf,
                                  unsigned* __restrict__ bar)
{
    const int i = blockIdx.x * blockDim.x + threadIdx.x;
    if (i == 0) *bar = 0u;
    if (i < BATCH * HID) {
        hbuf0[i] = h0[i];
        cbuf[i]  = c0[i];
    }
}

extern "C" void kernel_launch(void* const* d_in, const int* in_sizes, int n_in,
                              void* d_out, int out_size, void* d_ws, size_t ws_size,
                              hipStream_t stream) {
    const float* xs   = (const float*)d_in[0];
    const float* h0   = (const float*)d_in[1];
    const float* c0   = (const float*)d_in[2];
    const float* W_ih = (const float*)d_in[3];
    const float* W_hh = (const float*)d_in[4];
    const float* b_ih = (const float*)d_in[5];
    const float* b_hh = (const float*)d_in[6];
    float* out = (float*)d_out;
    float* ws  = (float*)d_ws;

    // Workspace layout: h ping-pong | c | barrier | (optional) gx
    const size_t BH = (size_t)BATCH * HID;
    float*    hbuf0 = ws;
    float*    hbuf1 = ws + BH;
    float*    cbuf  = ws + 2 * BH;
    unsigned* bar   = (unsigned*)(ws + 3 * BH);
    float*    gx    = ws + 3 * BH + 256;      // 1KB gap keeps gx aligned

    const size_t gx_elems  = (size_t)T_STEPS * BATCH * 4 * HID;       // 128M floats
    const size_t need      = (3 * BH + 256 + gx_elems) * sizeof(float);
    const bool   use_pre   = (ws_size >= need);

    init_state_kernel<<<((int)BH + 255) / 256, 256, 0, stream>>>(h0, c0, hbuf0, cbuf, bar);

    if (use_pre) {
        // 131072 waves: whole input projection in one parallel GEMM.
        input_gemm_kernel<<<16384, 256, 0, stream>>>(xs, W_ih, b_ih, b_hh, gx);
    }

    lstm_scan_kernel<<<SCAN_BLOCKS, 256, 0, stream>>>(
        xs, W_ih, W_hh, b_ih, b_hh, hbuf0, hbuf1, cbuf, out,
        use_pre ? gx : (const float*)nullptr, bar);
}